// FAIMHead_77678778515711
// MI455X (gfx1250) — compile-verified
//
#include <hip/hip_runtime.h>
#include <math.h>

#define EPS 1e-6f
#define B_N 128
#define C_N 1000
#define D_N 1024

typedef float v2f __attribute__((ext_vector_type(2)));
typedef float v8f __attribute__((ext_vector_type(8)));

#define WMMA_F32(a, b, c) \
  __builtin_amdgcn_wmma_f32_16x16x4_f32(false, (a), false, (b), (short)0, (c), false, false)

// ---------------------------------------------------------------------------
// Kernel 1: Lt = tril(L)
// ---------------------------------------------------------------------------
__global__ void build_tril_kernel(const float* __restrict__ L,
                                  float* __restrict__ Lt, int D) {
  int idx = blockIdx.x * blockDim.x + threadIdx.x;
  if (idx >= D * D) return;
  int i = idx / D;
  int j = idx - i * D;
  Lt[idx] = (j <= i) ? L[idx] : 0.0f;
}

// ---------------------------------------------------------------------------
// Kernel 2: Out[M x N] = A[M x K] @ Bm[K x N], row-major f32, K/N compile-time.
// One wave computes a 32x32 output patch (2x2 tiles of 16x16) with
// V_WMMA_F32_16X16X4_F32. Invalid M rows are CLAMPED (not zeroed): garbage
// accumulates only in output rows that the guarded store never writes, so the
// K-loop carries no EXEC masking at all.
// grid = (N/32, ceil(Mact/32)), block = 32 (one wave).
// ---------------------------------------------------------------------------
template <int K, int N>
__global__ __launch_bounds__(32) void gemm_f32_wmma(const float* __restrict__ A,
                                                    const float* __restrict__ Bm,
                                                    float* __restrict__ Out,
                                                    int Mact) {
  const int lane = threadIdx.x & 31;
  const int li = lane & 15;          // A: M row; B: N col (within 16-tile)
  const int kb = (lane >> 4) * 2;    // K sub-offset {0,2} within the 4-chunk
  const int n0 = blockIdx.x * 32;
  const int m0 = blockIdx.y * 32;

  int mA0 = m0 + li;
  int mA1 = m0 + 16 + li;
  if (mA0 >= Mact) mA0 = Mact - 1;   // clamp: affected D rows are never stored
  if (mA1 >= Mact) mA1 = Mact - 1;

  const float* Ap0 = A + (size_t)mA0 * K + kb;
  const float* Ap1 = A + (size_t)mA1 * K + kb;
  const float* Bp = Bm + (size_t)kb * N + n0 + li;

  v8f acc00 = {}, acc01 = {}, acc10 = {}, acc11 = {};

#pragma unroll 4
  for (int k0 = 0; k0 < K; k0 += 4) {
    v2f a0, a1, b0, b1;
    a0.x = Ap0[0];     a0.y = Ap0[1];       // one b64 load
    a1.x = Ap1[0];     a1.y = Ap1[1];       // one b64 load
    b0.x = Bp[0];      b0.y = Bp[N];        // immediate offsets
    b1.x = Bp[16];     b1.y = Bp[N + 16];
    acc00 = WMMA_F32(a0, b0, acc00);
    acc01 = WMMA_F32(a0, b1, acc01);
    acc10 = WMMA_F32(a1, b0, acc10);
    acc11 = WMMA_F32(a1, b1, acc11);
    Ap0 += 4;
    Ap1 += 4;
    Bp += 4 * N;
  }

  // D layout: lane li = N col; VGPR r holds M = r + 8*(lane>=16)
  const int rb = (lane >> 4) * 8;
#pragma unroll
  for (int r = 0; r < 8; ++r) {
    const int mmLo = m0 + rb + r;
    const int mmHi = m0 + 16 + rb + r;
    if (mmLo < Mact) {
      Out[(size_t)mmLo * N + n0 + li] = acc00[r];
      Out[(size_t)mmLo * N + n0 + 16 + li] = acc01[r];
    }
    if (mmHi < Mact) {
      Out[(size_t)mmHi * N + n0 + li] = acc10[r];
      Out[(size_t)mmHi * N + n0 + 16 + li] = acc11[r];
    }
  }
}

// ---------------------------------------------------------------------------
// Kernel 3: per-row stats: n2[r]=||R_r||^2, nT[r]=||T_r||^2, bd[r]=beta.R_r
// One wave per row.
// ---------------------------------------------------------------------------
__global__ __launch_bounds__(32) void row_stats_kernel(
    const float* __restrict__ R, const float* __restrict__ T,
    const float* __restrict__ beta, float* __restrict__ n2,
    float* __restrict__ nT, float* __restrict__ bd, int rows) {
  const int r = blockIdx.x;
  if (r >= rows) return;
  const int lane = threadIdx.x;
  float s2 = 0.0f, sT = 0.0f, sb = 0.0f;
  const float* Rr = R + (size_t)r * D_N;
  const float* Tr = T + (size_t)r * D_N;
#pragma unroll 4
  for (int d = lane; d < D_N; d += 32) {
    float rv = Rr[d];
    float tv = Tr[d];
    s2 += rv * rv;
    sT += tv * tv;
    sb += beta[d] * rv;
  }
#pragma unroll
  for (int off = 16; off > 0; off >>= 1) {
    s2 += __shfl_down(s2, off, 32);
    sT += __shfl_down(sT, off, 32);
    sb += __shfl_down(sb, off, 32);
  }
  if (lane == 0) {
    n2[r] = s2;
    nT[r] = sT;
    bd[r] = sb;
  }
}

// ---------------------------------------------------------------------------
// Kernel 4: fused cross-GEMMs + epilogue. One wave per 16x16 (b,c) tile.
//   acc1 = X' @ M'^T, acc2 = X @ Mu^T   (both K = D, contiguous b64 loads)
// out[b,c] = -scale*( sqrt(quad+EPS) + lmbda*sqrt(bdot^2+EPS) )
//   quad = (nxT[b]+nmT[c]-2*acc1) + EPS*(nx2[b]+nm2[c]-2*acc2)
// Invalid c rows are clamped; masked only at the store.
// grid = (ceil(C/16), B/16), block = 32.
// ---------------------------------------------------------------------------
__global__ __launch_bounds__(32) void cross_final_kernel(
    const float* __restrict__ X, const float* __restrict__ Mu,
    const float* __restrict__ XL, const float* __restrict__ ML,
    const float* __restrict__ nx2, const float* __restrict__ nxT,
    const float* __restrict__ bx, const float* __restrict__ nm2,
    const float* __restrict__ nmT, const float* __restrict__ bm,
    const float* __restrict__ lmbda_p, const float* __restrict__ scale_p,
    float* __restrict__ Out) {
  const int lane = threadIdx.x & 31;
  const int li = lane & 15;
  const int kb = (lane >> 4) * 2;
  const int c0 = blockIdx.x * 16;
  const int b0 = blockIdx.y * 16;

  const int brow = b0 + li;  // always < B_N (B_N % 16 == 0)
  int crow = c0 + li;
  if (crow >= C_N) crow = C_N - 1;  // clamp; masked at store

  const float* Xp = X + (size_t)brow * D_N + kb;
  const float* XLp = XL + (size_t)brow * D_N + kb;
  const float* Mup = Mu + (size_t)crow * D_N + kb;
  const float* MLp = ML + (size_t)crow * D_N + kb;

  v8f acc1 = {};
  v8f acc2 = {};

#pragma unroll 4
  for (int k0 = 0; k0 < D_N; k0 += 4) {
    v2f a1, a2, b1, b2;
    a1.x = XLp[0]; a1.y = XLp[1];
    b1.x = MLp[0]; b1.y = MLp[1];
    a2.x = Xp[0];  a2.y = Xp[1];
    b2.x = Mup[0]; b2.y = Mup[1];
    acc1 = WMMA_F32(a1, b1, acc1);
    acc2 = WMMA_F32(a2, b2, acc2);
    Xp += 4;
    XLp += 4;
    Mup += 4;
    MLp += 4;
  }

  const float lm = *lmbda_p;
  const float sc = *scale_p;
  const int c = c0 + li;
  const int rb = (lane >> 4) * 8;
  if (c < C_N) {
    const float nmTc = nmT[c];
    const float nm2c = nm2[c];
    const float bmc = bm[c];
#pragma unroll
    for (int r = 0; r < 8; ++r) {
      const int bb = b0 + rb + r;
      const float g1 = nxT[bb] + nmTc - 2.0f * acc1[r];
      const float g2 = nx2[bb] + nm2c - 2.0f * acc2[r];
      const float quad = g1 + EPS * g2;
      const float bdot = bx[bb] - bmc;
      const float val =
          -sc * (sqrtf(quad + EPS) + lm * sqrtf(bdot * bdot + EPS));
      Out[(size_t)bb * C_N + c] = val;
    }
  }
}

// ---------------------------------------------------------------------------
// Launch
// ---------------------------------------------------------------------------
extern "C" void kernel_launch(void* const* d_in, const int* in_sizes, int n_in,
                              void* d_out, int out_size, void* d_ws,
                              size_t ws_size, hipStream_t stream) {
  const float* x     = (const float*)d_in[0];  // [B, D]
  const float* mu    = (const float*)d_in[1];  // [C, D]
  const float* beta  = (const float*)d_in[2];  // [D]
  const float* L     = (const float*)d_in[3];  // [D, D]
  const float* lmbda = (const float*)d_in[4];  // scalar
  const float* scale = (const float*)d_in[5];  // scalar
  float* out = (float*)d_out;                  // [B, C]

  // Workspace layout (floats): Lt | XL | ML | row stats  (~8.8 MB total)
  float* ws = (float*)d_ws;
  float* Lt = ws;
  float* XL = Lt + (size_t)D_N * D_N;
  float* ML = XL + (size_t)B_N * D_N;
  float* st = ML + (size_t)C_N * D_N;
  float* nx2 = st;            // B_N
  float* nxT = nx2 + B_N;     // B_N
  float* bx  = nxT + B_N;     // B_N
  float* nm2 = bx + B_N;      // C_N
  float* nmT = nm2 + C_N;     // C_N
  float* bm  = nmT + C_N;     // C_N

  // 1) Lt = tril(L)
  build_tril_kernel<<<(D_N * D_N + 255) / 256, 256, 0, stream>>>(L, Lt, D_N);

  // 2) XL = X @ Lt   [128 x 1024]
  {
    dim3 grid(D_N / 32, B_N / 32);
    gemm_f32_wmma<D_N, D_N><<<grid, 32, 0, stream>>>(x, Lt, XL, B_N);
  }

  // 3) ML = Mu @ Lt  [1000 x 1024]  (row-clamped)
  {
    dim3 grid(D_N / 32, (C_N + 31) / 32);
    gemm_f32_wmma<D_N, D_N><<<grid, 32, 0, stream>>>(mu, Lt, ML, C_N);
  }

  // 4) row stats
  row_stats_kernel<<<B_N, 32, 0, stream>>>(x, XL, beta, nx2, nxT, bx, B_N);
  row_stats_kernel<<<C_N, 32, 0, stream>>>(mu, ML, beta, nm2, nmT, bm, C_N);

  // 5) fused cross-GEMMs + epilogue -> out
  {
    dim3 grid((C_N + 15) / 16, B_N / 16);
    cross_final_kernel<<<grid, 32, 0, stream>>>(x, mu, XL, ML, nx2, nxT, bx,
                                                nm2, nmT, bm, lmbda, scale,
                                                out);
  }
}